// GshardGate_79474074845410
// MI455X (gfx1250) — compile-verified
//
#include <hip/hip_runtime.h>
#include <hip/hip_bf16.h>
#include <math.h>

typedef __attribute__((ext_vector_type(2))) float v2f;
typedef __attribute__((ext_vector_type(4))) float v4f;
typedef __attribute__((ext_vector_type(8))) float v8f;

#define S_TOK 8192
#define MDIM  2048
#define NEXP  64
#define CAP   256
#define KC    128
#define LDA   (KC + 4)   // padded LDS stride: 16B-aligned rows, conflict-free b64 frag reads

// ---------------------------------------------------------------------------
// Kernel 1: logits[S,64] = inp[S,2048] @ wg[64,2048]^T  via V_WMMA_F32_16X16X4_F32
// 512 blocks of 128 threads (4 waves). Each wave: one 16x16 tile (16 tokens x 16 experts).
// ---------------------------------------------------------------------------
__global__ __launch_bounds__(128) void gate_gemm_wmma(
    const float* __restrict__ inp, const float* __restrict__ wg,
    float* __restrict__ logits)
{
  __shared__ float sA[16 * LDA];
  __shared__ float sB[NEXP * LDA];

  const int tid  = threadIdx.x;
  const int wave = tid >> 5;       // 0..3 -> expert column tile
  const int lane = tid & 31;
  const int l16  = lane & 15;
  const int half = lane >> 4;      // 0/1 -> K-pair (A/B) and M+8 (C/D)
  const int row0 = blockIdx.x * 16;
  const int e0   = wave * 16;

  v8f acc = {};

  for (int k0 = 0; k0 < MDIM; k0 += KC) {
    __syncthreads();
    // ---- cooperative load A chunk: 16 rows x KC floats (512 float4, 4/thread)
#pragma unroll
    for (int i = 0; i < 4; ++i) {
      int idx = tid + i * 128;
      int r = idx >> 5, c = idx & 31;
      v4f v = *(const v4f*)(inp + (size_t)(row0 + r) * MDIM + k0 + 4 * c);
      *(v4f*)(&sA[r * LDA + 4 * c]) = v;
    }
    // ---- cooperative load B chunk: 64 expert rows x KC floats (2048 float4, 16/thread)
#pragma unroll
    for (int i = 0; i < 16; ++i) {
      int idx = tid + i * 128;
      int r = idx >> 5, c = idx & 31;
      v4f v = *(const v4f*)(wg + (size_t)r * MDIM + k0 + 4 * c);
      *(v4f*)(&sB[r * LDA + 4 * c]) = v;
    }
    // hint the next A chunk into cache (global_prefetch_b8)
    if (k0 + KC < MDIM)
      __builtin_prefetch(inp + (size_t)(row0 + (tid >> 3)) * MDIM + k0 + KC, 0, 1);
    __syncthreads();

    // ---- 32 x WMMA f32 16x16x4 over this K-chunk
#pragma unroll
    for (int kk = 0; kk < KC / 4; ++kk) {
      // A frag: lane = row M=l16, elements K = 4*kk + 2*half + {0,1}
      v2f a = *(const v2f*)(&sA[l16 * LDA + kk * 4 + 2 * half]);
      // B frag: lane = col N=l16 (expert e0+l16), rows K = 4*kk + 2*half + {0,1}
      v2f b = *(const v2f*)(&sB[(e0 + l16) * LDA + kk * 4 + 2 * half]);
      acc = __builtin_amdgcn_wmma_f32_16x16x4_f32(
          /*neg_a=*/false, a, /*neg_b=*/false, b,
          /*c_mod=*/(short)0, acc, /*reuse_a=*/false, /*reuse_b=*/false);
    }
  }

  // C/D layout: VGPR v, lanes 0-15 -> M=v, lanes 16-31 -> M=v+8; N = l16
  const int m0 = row0 + half * 8;
#pragma unroll
  for (int v = 0; v < 8; ++v)
    logits[(size_t)(m0 + v) * NEXP + e0 + l16] = acc[v];
}

// ---------------------------------------------------------------------------
// Kernel 2: per-token softmax over 64 experts; one wave32 per token.
// Outputs argmax expert index and gate = 1 / sum(exp(l - max)).
// ---------------------------------------------------------------------------
__global__ __launch_bounds__(256) void gate_softmax(
    const float* __restrict__ logits, int* __restrict__ eidx,
    float* __restrict__ gatev)
{
  const int wave = threadIdx.x >> 5;
  const int lane = threadIdx.x & 31;
  const int s = blockIdx.x * 8 + wave;
  const float* lrow = logits + (size_t)s * NEXP;

  float a = lrow[lane];
  float b = lrow[lane + 32];

  float m  = a;
  int   mi = lane;
  if (b > m) { m = b; mi = lane + 32; }
  for (int off = 16; off; off >>= 1) {
    float om = __shfl_xor(m, off, 32);
    int   oi = __shfl_xor(mi, off, 32);
    if (om > m || (om == m && oi < mi)) { m = om; mi = oi; }  // first-max tie-break
  }
  float sum = expf(a - m) + expf(b - m);
  for (int off = 16; off; off >>= 1) sum += __shfl_xor(sum, off, 32);

  if (lane == 0) {
    eidx[s]  = mi;
    gatev[s] = 1.0f / sum;   // softmax value at the argmax
  }
}

// ---------------------------------------------------------------------------
// Kernel 3: one block per expert. Ordered scan over tokens assigned to this
// expert; slot = rank in token order; rank >= CAP -> dropped (-1).
// ---------------------------------------------------------------------------
__global__ __launch_bounds__(256) void gate_rank(
    const int* __restrict__ eidx, int* __restrict__ pos)
{
  const int e    = blockIdx.x;
  const int lane = threadIdx.x & 31;
  const int wave = threadIdx.x >> 5;
  __shared__ int wsum[8];
  __shared__ int sbase;
  if (threadIdx.x == 0) sbase = 0;
  __syncthreads();

  for (int chunk = 0; chunk < S_TOK; chunk += 256) {
    const int s = chunk + threadIdx.x;
    const bool f = (eidx[s] == e);
    unsigned mask = (unsigned)__ballot(f);            // wave32: low 32 bits
    const int wprefix = __popc(mask & ((1u << lane) - 1u));
    const int wtotal  = __popc(mask);
    if (lane == 0) wsum[wave] = wtotal;
    __syncthreads();
    int waveoff = 0, total = 0;
    for (int w = 0; w < 8; ++w) {
      if (w < wave) waveoff += wsum[w];
      total += wsum[w];
    }
    const int p = sbase + waveoff + wprefix;
    if (f) pos[s] = (p < CAP) ? p : -1;
    __syncthreads();
    if (threadIdx.x == 0) sbase += total;
    __syncthreads();
  }
}

// ---------------------------------------------------------------------------
// Kernel 4: stream 1 GiB of zeros into d_out with nontemporal float4 stores.
// ---------------------------------------------------------------------------
__global__ __launch_bounds__(256) void fill_zero_f4(v4f* __restrict__ out, long n4)
{
  const long stride = (long)gridDim.x * blockDim.x;
  v4f z = {};
  for (long i = (long)blockIdx.x * blockDim.x + threadIdx.x; i < n4; i += stride)
    __builtin_nontemporal_store(z, &out[i]);
}

// ---------------------------------------------------------------------------
// Kernel 5: scatter the one-hot values: combine[s,e,c]=gate, dispatch[s,e,c]=1.
// ---------------------------------------------------------------------------
__global__ __launch_bounds__(256) void gate_scatter(
    const int* __restrict__ eidx, const float* __restrict__ gatev,
    const int* __restrict__ pos, float* __restrict__ out)
{
  const int s = blockIdx.x * blockDim.x + threadIdx.x;
  if (s >= S_TOK) return;
  const int p = pos[s];
  if (p >= 0) {
    const int e = eidx[s];
    const size_t off = ((size_t)s * NEXP + e) * CAP + p;
    out[off] = gatev[s];                                   // combine_weights
    out[(size_t)S_TOK * NEXP * CAP + off] = 1.0f;          // dispatch_mask
  }
}

// ---------------------------------------------------------------------------
extern "C" void kernel_launch(void* const* d_in, const int* in_sizes, int n_in,
                              void* d_out, int out_size, void* d_ws, size_t ws_size,
                              hipStream_t stream) {
  (void)in_sizes; (void)n_in; (void)ws_size;
  const float* inp = (const float*)d_in[0];   // [8192, 2048] fp32
  const float* wg  = (const float*)d_in[1];   // [64, 2048]   fp32

  char* ws = (char*)d_ws;
  float* logits = (float*)ws;                                    // 8192*64 f32
  int*   eidx   = (int*)(ws + (size_t)S_TOK * NEXP * 4);         // 8192 i32
  float* gatev  = (float*)(ws + (size_t)S_TOK * NEXP * 4 + S_TOK * 4);
  int*   pos    = (int*)(ws + (size_t)S_TOK * NEXP * 4 + 2 * S_TOK * 4);

  gate_gemm_wmma<<<S_TOK / 16, 128, 0, stream>>>(inp, wg, logits);
  gate_softmax<<<S_TOK / 8, 256, 0, stream>>>(logits, eidx, gatev);
  gate_rank<<<NEXP, 256, 0, stream>>>(eidx, pos);

  const long n4 = (long)out_size / 4;          // out_size = 2*8192*64*256, divisible by 4
  fill_zero_f4<<<8192, 256, 0, stream>>>((v4f*)d_out, n4);
  gate_scatter<<<S_TOK / 256, 256, 0, stream>>>(eidx, gatev, pos, (float*)d_out);
}